// DeepGraphConv_Surv_module_68642167325075
// MI455X (gfx1250) — compile-verified
//
#include <hip/hip_runtime.h>
#include <hip/hip_bf16.h>
#include <stdint.h>

#define N_NODES 50000
#define HID 128
#define N_EDGES 600000
#define LSTRIDE 132   // LDS row stride in floats: 8B-aligned rows, 4-bank rotation -> conflict-free b64 loads

typedef __attribute__((ext_vector_type(2))) float v2f;
typedef __attribute__((ext_vector_type(8))) float v8f;

// ---------------------------------------------------------------------------
// 1) Dropout mask: replicate jax.random.bernoulli(jax.random.key(42), 0.75, (128,))
//    threefry2x32, key = [0, 42], counts = iota(128) split into (i, i+64) pairs.
// ---------------------------------------------------------------------------
__device__ __forceinline__ uint32_t rotl32(uint32_t v, int r) {
  return (v << r) | (v >> (32 - r));
}

__global__ void gin_mask_kernel(float* __restrict__ mask) {
  int i = threadIdx.x;  // 0..127
  uint32_t lo = (i < 64) ? (uint32_t)i : (uint32_t)(i - 64);
  uint32_t x0 = lo, x1 = lo + 64u;
  const uint32_t k0 = 0u, k1 = 42u, k2 = 0u ^ 42u ^ 0x1BD11BDAu;
  x0 += k0; x1 += k1;
#define TF_R(r) { x0 += x1; x1 = rotl32(x1, (r)); x1 ^= x0; }
#define TF_G0 TF_R(13) TF_R(15) TF_R(26) TF_R(6)
#define TF_G1 TF_R(17) TF_R(29) TF_R(16) TF_R(24)
  TF_G0; x0 += k1; x1 += k2 + 1u;
  TF_G1; x0 += k2; x1 += k0 + 2u;
  TF_G0; x0 += k0; x1 += k1 + 3u;
  TF_G1; x0 += k1; x1 += k2 + 4u;
  TF_G0; x0 += k2; x1 += k0 + 5u;
#undef TF_G1
#undef TF_G0
#undef TF_R
  uint32_t bits = (i < 64) ? x0 : x1;
  float u = __uint_as_float((bits >> 9) | 0x3f800000u) - 1.0f;  // U[0,1)
  mask[i] = (u < 0.75f) ? 1.0f : 0.0f;
}

// ---------------------------------------------------------------------------
// 2) Zero the aggregation buffer (which aliases d_out).
// ---------------------------------------------------------------------------
__global__ void gin_zero_kernel(float4* __restrict__ p, int n4) {
  int i = blockIdx.x * blockDim.x + threadIdx.x;
  int stride = gridDim.x * blockDim.x;
  float4 z; z.x = 0.f; z.y = 0.f; z.z = 0.f; z.w = 0.f;
  for (; i < n4; i += stride) p[i] = z;
}

// ---------------------------------------------------------------------------
// 3) Edge scatter: agg[dst] += x[src] * mask. One wave32 per edge, float4/lane.
// ---------------------------------------------------------------------------
__global__ __launch_bounds__(256) void gin_scatter_kernel(
    const long long* __restrict__ ei, const float* __restrict__ x,
    const float* __restrict__ mask, float* __restrict__ agg) {
  const int lane = threadIdx.x & 31;
  const long long e = (long long)blockIdx.x * 8 + (threadIdx.x >> 5);
  if (e >= N_EDGES) return;
  const long long src = ei[e];
  const long long dst = ei[N_EDGES + e];
  const float4 m = ((const float4*)mask)[lane];
  const float4 v = ((const float4*)(x + src * HID))[lane];
  float* d = agg + dst * HID + lane * 4;
  atomicAdd(d + 0, v.x * m.x);
  atomicAdd(d + 1, v.y * m.y);
  atomicAdd(d + 2, v.z * m.z);
  atomicAdd(d + 3, v.w * m.w);
}

// ---------------------------------------------------------------------------
// 4) Fused MLP: h = x*mask + agg; h1 = relu(h@W1+b1); out = h1@W2+b2.
//    8 waves/block, 16 node-rows per wave, V_WMMA_F32_16X16X4_F32.
// ---------------------------------------------------------------------------
__global__ __launch_bounds__(256) void gin_mlp_wmma_kernel(
    const float* __restrict__ x, const float* __restrict__ agg,
    const float* __restrict__ mask,
    const float* __restrict__ W1, const float* __restrict__ b1,
    const float* __restrict__ W2, const float* __restrict__ b2,
    float* __restrict__ out) {
  __shared__ float Wt[HID * LSTRIDE];  // Wt[j*LSTRIDE + k] = W[k][j]
  __shared__ float Ht[HID * LSTRIDE];  // Ht[r*LSTRIDE + k]

  const int tid = threadIdx.x;
  const int wid = tid >> 5;
  const int lane = tid & 31;
  const int Lm = lane & 15;
  const int hi = lane >> 4;
  const int rowBase = blockIdx.x * HID;
  const int r0 = wid * 16;

  // ---- stage H = x*mask + agg into LDS (each wave loads rows wid, wid+8, ...)
  {
    const float4 m4 = ((const float4*)mask)[lane];
    for (int rr = wid; rr < HID; rr += 8) {
      int g = rowBase + rr;
      if (g > N_NODES - 1) g = N_NODES - 1;  // clamp tail (stores are guarded)
      const float4 xv = ((const float4*)(x + (long long)g * HID))[lane];
      const float4 av = ((const float4*)(agg + (long long)g * HID))[lane];
      float4 h;
      h.x = xv.x * m4.x + av.x;
      h.y = xv.y * m4.y + av.y;
      h.z = xv.z * m4.z + av.z;
      h.w = xv.w * m4.w + av.w;
      *((float4*)&Ht[rr * LSTRIDE + lane * 4]) = h;
    }
  }
  // ---- stage W1^T
  for (int idx = tid; idx < HID * HID; idx += 256) {
    int k = idx >> 7, j = idx & 127;
    Wt[j * LSTRIDE + k] = W1[idx];
  }
  __syncthreads();

  const float* aRow = &Ht[(r0 + Lm) * LSTRIDE + 2 * hi];

  // ---- layer 1: acc[c] = Htile(16xK) @ W1tile(Kx16), K = 128
  v8f acc[8];
#pragma unroll
  for (int c = 0; c < 8; ++c) {
#pragma unroll
    for (int v = 0; v < 8; ++v) acc[c][v] = 0.0f;
  }
  for (int k0 = 0; k0 < HID; k0 += 4) {
    v2f a = *(const v2f*)(aRow + k0);
#pragma unroll
    for (int c = 0; c < 8; ++c) {
      v2f b = *(const v2f*)&Wt[(c * 16 + Lm) * LSTRIDE + k0 + 2 * hi];
      acc[c] = __builtin_amdgcn_wmma_f32_16x16x4_f32(
          false, a, false, b, (short)0, acc[c], false, false);
    }
  }
  __syncthreads();  // all waves done reading Wt(W1) before it is overwritten

  // ---- bias + ReLU, write h1 back into this wave's own Ht rows
#pragma unroll
  for (int c = 0; c < 8; ++c) {
    float bb = b1[c * 16 + Lm];
#pragma unroll
    for (int v = 0; v < 8; ++v) {
      float val = fmaxf(acc[c][v] + bb, 0.0f);
      Ht[(r0 + v + 8 * hi) * LSTRIDE + c * 16 + Lm] = val;
    }
  }
  // ---- stage W2^T into the same LDS buffer
  for (int idx = tid; idx < HID * HID; idx += 256) {
    int k = idx >> 7, j = idx & 127;
    Wt[j * LSTRIDE + k] = W2[idx];
  }
  __syncthreads();

  // ---- layer 2
  v8f acc2[8];
#pragma unroll
  for (int c = 0; c < 8; ++c) {
#pragma unroll
    for (int v = 0; v < 8; ++v) acc2[c][v] = 0.0f;
  }
  for (int k0 = 0; k0 < HID; k0 += 4) {
    v2f a = *(const v2f*)(aRow + k0);
#pragma unroll
    for (int c = 0; c < 8; ++c) {
      v2f b = *(const v2f*)&Wt[(c * 16 + Lm) * LSTRIDE + k0 + 2 * hi];
      acc2[c] = __builtin_amdgcn_wmma_f32_16x16x4_f32(
          false, a, false, b, (short)0, acc2[c], false, false);
    }
  }

  // ---- epilogue: bias2, guarded global store
#pragma unroll
  for (int c = 0; c < 8; ++c) {
    float bb = b2[c * 16 + Lm];
#pragma unroll
    for (int v = 0; v < 8; ++v) {
      int g = rowBase + r0 + v + 8 * hi;
      if (g < N_NODES) out[(long long)g * HID + c * 16 + Lm] = acc2[c][v] + bb;
    }
  }
}

// ---------------------------------------------------------------------------
extern "C" void kernel_launch(void* const* d_in, const int* in_sizes, int n_in,
                              void* d_out, int out_size, void* d_ws, size_t ws_size,
                              hipStream_t stream) {
  const float* x = (const float*)d_in[0];
  const long long* ei = (const long long*)d_in[1];
  const float* W1 = (const float*)d_in[2];
  const float* b1 = (const float*)d_in[3];
  const float* W2 = (const float*)d_in[4];
  const float* b2 = (const float*)d_in[5];
  float* out = (float*)d_out;
  float* mask = (float*)d_ws;   // 128 floats
  float* agg = out;             // aggregate into d_out; consumed before overwrite

  gin_mask_kernel<<<1, 128, 0, stream>>>(mask);

  const int n4 = N_NODES * HID / 4;
  gin_zero_kernel<<<2048, 256, 0, stream>>>((float4*)agg, n4);

  gin_scatter_kernel<<<(N_EDGES + 7) / 8, 256, 0, stream>>>(ei, x, mask, agg);

  gin_mlp_wmma_kernel<<<(N_NODES + HID - 1) / HID, 256, 0, stream>>>(
      x, agg, mask, W1, b1, W2, b2, out);
}